// TemporalGNN_47201690583389
// MI455X (gfx1250) — compile-verified
//
#include <hip/hip_runtime.h>
#include <math.h>

#define NN 50000
#define EE 500000
#define TF 96           // T*F = 12*8 aggregated channels per node

typedef __attribute__((ext_vector_type(2))) float v2f;
typedef __attribute__((ext_vector_type(8))) float v8f;

__device__ __forceinline__ v8f wmma4(v2f a, v2f b, v8f c) {
  // D(16x16,f32) = A(16x4,f32) * B(4x16,f32) + C
  return __builtin_amdgcn_wmma_f32_16x16x4_f32(false, a, false, b, (short)0, c,
                                               false, false);
}

// ---------------------------------------------------------------- init
__global__ void k_init(float* __restrict__ Y, float* __restrict__ deg) {
  int i = blockIdx.x * blockDim.x + threadIdx.x;
  int stride = gridDim.x * blockDim.x;
  for (int j = i; j < NN * TF; j += stride) Y[j] = 0.0f;
  for (int j = i; j < NN; j += stride) deg[j] = 1.0f;  // self-loop weight
}

// ------------------------------------------- fold weights + softmax(att)
// W2[f*64+c] : c<32 -> conv_z_W@lin_z_W[:32], c>=32 -> conv_h_W@lin_h_W[:32]
__global__ void k_prep(const float* __restrict__ czW, const float* __restrict__ czb,
                       const float* __restrict__ chW, const float* __restrict__ chb,
                       const float* __restrict__ lzW, const float* __restrict__ lzb,
                       const float* __restrict__ lhW, const float* __restrict__ lhb,
                       const float* __restrict__ att,
                       float* __restrict__ W2, float* __restrict__ bcomb,
                       float* __restrict__ probs) {
  int c = threadIdx.x;                 // 0..63
  const float* convW = (c < 32) ? czW : chW;
  const float* convb = (c < 32) ? czb : chb;
  const float* L     = (c < 32) ? lzW : lhW;
  const float* lb    = (c < 32) ? lzb : lhb;
  int cc = c & 31;
  for (int f = 0; f < 8; ++f) {
    float s = 0.0f;
    for (int k = 0; k < 32; ++k) s += convW[f * 32 + k] * L[k * 32 + cc];
    W2[f * 64 + c] = s;
  }
  float bs = lb[cc];
  for (int k = 0; k < 32; ++k) bs += convb[k] * L[k * 32 + cc];
  bcomb[c] = bs;
  if (c == 0) {
    float mx = -1e30f;
    for (int t = 0; t < 12; ++t) mx = fmaxf(mx, att[t]);
    float ex[12], sum = 0.0f;
    for (int t = 0; t < 12; ++t) { ex[t] = expf(att[t] - mx); sum += ex[t]; }
    for (int t = 0; t < 12; ++t) probs[t] = ex[t] / sum;
  }
}

// ------------------------------------- edge weight MLP + degree atomics
__global__ void k_edge(const float* __restrict__ ef, const float* __restrict__ efW,
                       const float* __restrict__ efb, const int* __restrict__ ei,
                       float* __restrict__ ew, float* __restrict__ deg) {
  int e = blockIdx.x * blockDim.x + threadIdx.x;
  if (e >= EE) return;
  float4 f = *(const float4*)(ef + (size_t)e * 4);
  float w = f.x * efW[0] + f.y * efW[1] + f.z * efW[2] + f.w * efW[3] + efb[0];
  w = fmaxf(w, 0.0f);
  ew[e] = w;
  unsafeAtomicAdd(&deg[ei[EE + e]], w);      // col = targets
}

__global__ void k_dinv(float* __restrict__ deg) {
  int i = blockIdx.x * blockDim.x + threadIdx.x;
  if (i < NN) deg[i] = rsqrtf(deg[i]);       // deg >= 1 always (relu + self loop)
}

// ------------------------- feature linear: [N*T x 8] @ feat_W[8x8] (WMMA)
// x[n][g][t] at n*96+g*12+t ; xt[n][t*8+f] = row r=n*12+t -> xt[r*8+f]
__global__ void __launch_bounds__(128) k_featlin(const float* __restrict__ x,
                                                 const float* __restrict__ fW,
                                                 const float* __restrict__ fb,
                                                 float* __restrict__ xt) {
  const int M = NN * 12;
  int wave = (blockIdx.x * blockDim.x + threadIdx.x) >> 5;
  int lane = threadIdx.x & 31;
  int r0 = wave * 16;
  if (r0 >= M) return;                       // wave-uniform
  int half = lane >> 4, lr = lane & 15;
  int row = r0 + lr;
  int rowc = (row < M) ? row : (M - 1);
  int n = rowc / 12, t = rowc % 12;
  const float* xr = x + (size_t)n * 96 + t;  // element g at xr[g*12]
  v8f acc = {};
#pragma unroll
  for (int s = 0; s < 2; ++s) {
    int k0 = 4 * s + 2 * half;
    v2f a; a.x = xr[k0 * 12]; a.y = xr[(k0 + 1) * 12];
    v2f b;
    b.x = (lr < 8) ? fW[k0 * 8 + lr] : 0.0f;
    b.y = (lr < 8) ? fW[(k0 + 1) * 8 + lr] : 0.0f;
    acc = wmma4(a, b, acc);
  }
  float bias = (lr < 8) ? fb[lr] : 0.0f;
#pragma unroll
  for (int j = 0; j < 8; ++j) {
    int orow = r0 + half * 8 + j;
    if (orow < M && lr < 8) xt[(size_t)orow * 8 + lr] = acc[j] + bias;
  }
}

// ------------- sparse aggregation Y[dst] += norm*xt[src]  (8 lanes/edge)
// lane group of 8 covers the contiguous 96 floats of one edge -> coalesced
// gathers and coalesced atomic segments.
__global__ void k_scatter(const float* __restrict__ xt, const float* __restrict__ ew,
                          const float* __restrict__ dinv, const int* __restrict__ ei,
                          float* __restrict__ Y) {
  long long tid = (long long)blockIdx.x * blockDim.x + threadIdx.x;
  int e = (int)(tid >> 3);
  int p = (int)(tid & 7);                    // part: floats [p*12, p*12+12)
  if (e >= EE) return;
  int src = ei[e], dst = ei[EE + e];
  float nrm = dinv[src] * ew[e] * dinv[dst];
  if (nrm == 0.0f) return;
  const float* xs = xt + (size_t)src * TF + p * 12;
  float* yd = Y + (size_t)dst * TF + p * 12;
  float4 v0 = *(const float4*)(xs);
  float4 v1 = *(const float4*)(xs + 4);
  float4 v2 = *(const float4*)(xs + 8);
  unsafeAtomicAdd(&yd[0],  nrm * v0.x);
  unsafeAtomicAdd(&yd[1],  nrm * v0.y);
  unsafeAtomicAdd(&yd[2],  nrm * v0.z);
  unsafeAtomicAdd(&yd[3],  nrm * v0.w);
  unsafeAtomicAdd(&yd[4],  nrm * v1.x);
  unsafeAtomicAdd(&yd[5],  nrm * v1.y);
  unsafeAtomicAdd(&yd[6],  nrm * v1.z);
  unsafeAtomicAdd(&yd[7],  nrm * v1.w);
  unsafeAtomicAdd(&yd[8],  nrm * v2.x);
  unsafeAtomicAdd(&yd[9],  nrm * v2.y);
  unsafeAtomicAdd(&yd[10], nrm * v2.z);
  unsafeAtomicAdd(&yd[11], nrm * v2.w);
}

// -------- fused: gates (WMMA) + attention-accumulate + relu + head (WMMA)
__global__ void __launch_bounds__(128) k_main(const float* __restrict__ Y,
                                              const float* __restrict__ xt,
                                              const float* __restrict__ dinv,
                                              const float* __restrict__ W2g,
                                              const float* __restrict__ bg,
                                              const float* __restrict__ pg,
                                              const float* __restrict__ headW,
                                              const float* __restrict__ headb,
                                              float* __restrict__ out) {
  __shared__ float sW2[8 * 64];
  __shared__ float sB[64];
  __shared__ float sP[12];
  __shared__ float hbuf[4][16][32];

  for (int i = threadIdx.x; i < 512; i += 128) sW2[i] = W2g[i];
  if (threadIdx.x < 64) sB[threadIdx.x] = bg[threadIdx.x];
  if (threadIdx.x < 12) sP[threadIdx.x] = pg[threadIdx.x];
  __syncthreads();

  int w = threadIdx.x >> 5;
  int lane = threadIdx.x & 31;
  int half = lane >> 4, lr = lane & 15;
  int tile = blockIdx.x * 4 + w;
  int base = tile * 16;
  int na = base + lr; if (na >= NN) na = NN - 1;   // clamped A-row node
  float dv = dinv[na];
  float selfw = dv * dv;                            // self-loop norm

  // B tiles for the fused [8 x 64] gate weight (t-invariant)
  v2f b[4][2];
#pragma unroll
  for (int c = 0; c < 4; ++c)
#pragma unroll
    for (int s = 0; s < 2; ++s) {
      int k0 = 4 * s + 2 * half, col = 16 * c + lr;
      b[c][s].x = sW2[k0 * 64 + col];
      b[c][s].y = sW2[(k0 + 1) * 64 + col];
    }

  const int kb = 2 * half;
  v8f acc0 = {}, acc1 = {};
  for (int t = 0; t < 12; ++t) {
    size_t off = (size_t)na * TF + t * 8;
    // vectorized b64 loads: (off + kb) and (off + 4 + kb) are 2-float aligned
    float2 y0 = *(const float2*)(Y + off + kb);
    float2 x0 = *(const float2*)(xt + off + kb);
    float2 y1 = *(const float2*)(Y + off + 4 + kb);
    float2 x1 = *(const float2*)(xt + off + 4 + kb);
    v2f a0, a1;
    a0.x = y0.x + selfw * x0.x;  a0.y = y0.y + selfw * x0.y;
    a1.x = y1.x + selfw * x1.x;  a1.y = y1.y + selfw * x1.y;
    v8f d0 = {}, d1 = {}, d2 = {}, d3 = {};
    d0 = wmma4(a0, b[0][0], d0); d0 = wmma4(a1, b[0][1], d0);
    d1 = wmma4(a0, b[1][0], d1); d1 = wmma4(a1, b[1][1], d1);
    d2 = wmma4(a0, b[2][0], d2); d2 = wmma4(a1, b[2][1], d2);
    d3 = wmma4(a0, b[3][0], d3); d3 = wmma4(a1, b[3][1], d3);
    float p = sP[t];
#pragma unroll
    for (int j = 0; j < 8; ++j) {
      float z0 = 1.0f / (1.0f + expf(-(d0[j] + sB[lr])));
      float z1 = 1.0f / (1.0f + expf(-(d1[j] + sB[16 + lr])));
      float h0 = tanhf(d2[j] + sB[32 + lr]);
      float h1 = tanhf(d3[j] + sB[48 + lr]);
      acc0[j] += p * (1.0f - z0) * h0;   // H=0 => cell = (1-Z)*H_tilde
      acc1[j] += p * (1.0f - z1) * h1;
    }
  }

  // relu -> LDS (C-layout -> A-layout transpose for the head GEMM)
  float* hb = &hbuf[w][0][0];
#pragma unroll
  for (int j = 0; j < 8; ++j) {
    int r = half * 8 + j;
    hb[r * 32 + lr]      = fmaxf(acc0[j], 0.0f);
    hb[r * 32 + 16 + lr] = fmaxf(acc1[j], 0.0f);
  }
  __syncthreads();

  // head: [16x32] @ head_W[32x12 pad 16]
  v8f dh = {};
#pragma unroll
  for (int s = 0; s < 8; ++s) {
    int k0 = 4 * s + 2 * half;
    v2f a; a.x = hb[lr * 32 + k0]; a.y = hb[lr * 32 + k0 + 1];
    v2f bb;
    bb.x = (lr < 12) ? headW[k0 * 12 + lr] : 0.0f;
    bb.y = (lr < 12) ? headW[(k0 + 1) * 12 + lr] : 0.0f;
    dh = wmma4(a, bb, dh);
  }
  float hbias = (lr < 12) ? headb[lr] : 0.0f;
#pragma unroll
  for (int j = 0; j < 8; ++j) {
    int node = base + half * 8 + j;
    if (node < NN && lr < 12) out[(size_t)node * 12 + lr] = dh[j] + hbias;
  }
}

// ---------------------------------------------------------------- launch
extern "C" void kernel_launch(void* const* d_in, const int* in_sizes, int n_in,
                              void* d_out, int out_size, void* d_ws, size_t ws_size,
                              hipStream_t stream) {
  const float* x   = (const float*)d_in[0];
  const int*   ei  = (const int*)d_in[1];
  const float* ef  = (const float*)d_in[2];
  const float* fW  = (const float*)d_in[3];
  const float* fb  = (const float*)d_in[4];
  const float* efW = (const float*)d_in[5];
  const float* efb = (const float*)d_in[6];
  const float* att = (const float*)d_in[7];
  const float* czW = (const float*)d_in[8];
  const float* czb = (const float*)d_in[9];
  // d_in[10], d_in[11]: conv_r_* (dead: H=0 -> H*R=0)
  const float* chW = (const float*)d_in[12];
  const float* chb = (const float*)d_in[13];
  const float* lzW = (const float*)d_in[14];
  const float* lzb = (const float*)d_in[15];
  // d_in[16], d_in[17]: lin_r_* (dead)
  const float* lhW = (const float*)d_in[18];
  const float* lhb = (const float*)d_in[19];
  const float* hW  = (const float*)d_in[20];
  const float* hb  = (const float*)d_in[21];
  float* out = (float*)d_out;

  float* ws    = (float*)d_ws;
  float* ew    = ws;                         // E
  float* dinv  = ew + EE;                    // N (deg, then rsqrt in place)
  float* xt    = dinv + NN;                  // N*96
  float* Y     = xt + (size_t)NN * TF;       // N*96
  float* W2    = Y + (size_t)NN * TF;        // 8*64
  float* bcomb = W2 + 512;                   // 64
  float* probs = bcomb + 64;                 // 12

  k_init<<<4096, 256, 0, stream>>>(Y, dinv);
  k_prep<<<1, 64, 0, stream>>>(czW, czb, chW, chb, lzW, lzb, lhW, lhb, att,
                               W2, bcomb, probs);
  k_edge<<<(EE + 255) / 256, 256, 0, stream>>>(ef, efW, efb, ei, ew, dinv);
  k_dinv<<<(NN + 255) / 256, 256, 0, stream>>>(dinv);
  k_featlin<<<(NN * 12 / 16 + 3) / 4, 128, 0, stream>>>(x, fW, fb, xt);
  k_scatter<<<(int)(((long long)EE * 8 + 255) / 256), 256, 0, stream>>>(xt, ew,
                                                                        dinv, ei, Y);
  k_main<<<((NN + 15) / 16 + 3) / 4, 128, 0, stream>>>(Y, xt, dinv, W2, bcomb,
                                                       probs, hW, hb, out);
}